// PointNetModule_6347961663562
// MI455X (gfx1250) — compile-verified
//
#include <hip/hip_runtime.h>

// ---- problem constants (match reference) ----
#define BB     16
#define NPTS   2048
#define PCTR   256
#define KNB    32
#define HW     (24*80)
#define NFEA   64
#define MDIM   (BB*PCTR*KNB)   // 131072 GEMM columns (power of two, 2^17)
#define DISTC  0.5f
#define EPSC   1e-5f

typedef float v2f __attribute__((ext_vector_type(2)));
typedef float v8f __attribute__((ext_vector_type(8)));

// X buffers use channel-pair interleaved layout:
//   X[(c>>1) * 2*MDIM + m*2 + (c&1)]
// so a WMMA B-fragment {c,c+1} (or {c+2,c+3} for lanes 16-31) is one aligned
// 8-byte load, and a wave's 16 lanes read 128B dense.

// ---------------------------------------------------------------------------
// 1) query_depth_point: one thread per (b, center). First K indices with
//    |dz| < DIST in index order; pad with first hit (0 if none); valid = num>0.
// ---------------------------------------------------------------------------
__global__ void k_query(const float* __restrict__ pc, const float* __restrict__ new_pc,
                        int* __restrict__ idx, float* __restrict__ validf) {
  int t = blockIdx.x * blockDim.x + threadIdx.x;
  if (t >= BB * PCTR) return;
  int b = t / PCTR, p = t % PCTR;
  float zc = new_pc[(b * 3 + 2) * PCTR + p];
  const float* z = pc + (size_t)(b * 3 + 2) * NPTS;
  int* row = idx + (size_t)t * KNB;
  int cnt = 0, first = 0;
  for (int i = 0; i < NPTS && cnt < KNB; ++i) {
    if (fabsf(z[i] - zc) < DISTC) {
      if (cnt == 0) first = i;
      row[cnt++] = i;
    }
  }
  for (int j = cnt; j < KNB; ++j) row[j] = first;
  validf[t] = (cnt > 0) ? 1.0f : 0.0f;
}

// ---------------------------------------------------------------------------
// 1b) zero-pad w1 (64 x 67) -> Wpad (64 x 68)
// ---------------------------------------------------------------------------
__global__ void k_padw1(const float* __restrict__ w, float* __restrict__ wp) {
  int t = blockIdx.x * blockDim.x + threadIdx.x;
  if (t >= 64 * 68) return;
  int o = t / 68, c = t % 68;
  wp[t] = (c < 67) ? w[o * 67 + c] : 0.0f;
}

// ---------------------------------------------------------------------------
// 2) gather: build X0 (68 ch, pair-interleaved; ch67 zero pad) and write the
//    rgb1/rgb2 output slices (masked). One thread per column m.
// ---------------------------------------------------------------------------
__global__ void k_gather(const float* __restrict__ pc, const float* __restrict__ feat,
                         const float* __restrict__ img1, const float* __restrict__ img2,
                         const int* __restrict__ qv1, const float* __restrict__ new_pc,
                         const int* __restrict__ idx, const float* __restrict__ validf,
                         float* __restrict__ X0, float* __restrict__ out) {
  int m = blockIdx.x * blockDim.x + threadIdx.x;
  if (m >= MDIM) return;
  int b = m >> 13;          // 8192 columns per batch
  int rem = m & 8191;       // p*32 + k
  int p = rem >> 5;
  int i = idx[m];

  float arr[68];
#pragma unroll
  for (int c = 0; c < 3; ++c)
    arr[c] = pc[(size_t)(b * 3 + c) * NPTS + i] - new_pc[(size_t)(b * 3 + c) * PCTR + p];
#pragma unroll
  for (int c = 0; c < NFEA; ++c)
    arr[3 + c] = feat[(size_t)(b * NFEA + c) * NPTS + i];
  arr[67] = 0.0f;
#pragma unroll
  for (int pr = 0; pr < 34; ++pr) {
    v2f w2 = {arr[2 * pr], arr[2 * pr + 1]};
    *(v2f*)(X0 + (size_t)pr * 2 * MDIM + (size_t)m * 2) = w2;
  }

  int q = qv1[(size_t)b * NPTS + i];
  float vm = validf[(b << 8) + p];
  float* ob = out + (size_t)b * 640 * 8192 + rem;
  for (int c = 0; c < 64; ++c)      // f1 rgb slice -> channels 64..127
    ob[(size_t)(64 + c) * 8192] = img1[(size_t)(b * 64 + c) * HW + q] * vm;
  for (int c = 0; c < 128; ++c)     // f2 rgb slice -> channels 256..383
    ob[(size_t)(256 + c) * 8192] = img2[(size_t)(b * 128 + c) * HW + q] * vm;
}

// ---------------------------------------------------------------------------
// 3) GEMM via V_WMMA_F32_16X16X4_F32. One 16(out) x 64(m) strip per wave:
//    4 accumulators share each A fragment, k-loop over CIN (compile-time)
//    in steps of 4, all A/B loads are single b64. Grid sized exactly ->
//    EXEC all ones as WMMA requires.
//    A(16x4): lanes 0-15 hold {c,c+1} of row M=lane&15; lanes 16-31 {c+2,c+3}.
//    B(4x16) mirrored. C/D: VGPR r -> M=r (lanes 0-15) / r+8, N=lane&15.
// ---------------------------------------------------------------------------
template <int CIN>
__global__ void k_gemm(const float* __restrict__ W, const float* __restrict__ Xi,
                       float* __restrict__ Y) {
  int wave = blockIdx.x * (blockDim.x >> 5) + (threadIdx.x >> 5);
  int lane = threadIdx.x & 31;
  const int strips_m = MDIM / 64;   // 2048
  int ot = wave / strips_m;
  int ms = wave % strips_m;
  int hi = lane >> 4;
  int l4 = lane & 15;

  const float* wr = W + (size_t)(ot * 16 + l4) * CIN + (hi << 1);
  // interleaved X base for this lane: column (ms*64 + l4), pair bump for hi lanes
  const float* xb = Xi + (size_t)(ms * 64 + l4) * 2 + (size_t)(hi ? 2 * MDIM : 0);

  v8f acc0 = {0.f,0.f,0.f,0.f,0.f,0.f,0.f,0.f};
  v8f acc1 = acc0, acc2 = acc0, acc3 = acc0;

#pragma unroll 4
  for (int c = 0; c < CIN; c += 4) {
    v2f a  = *(const v2f*)(wr + c);                      // b64: W{c,c+1} (+2 for hi)
    const float* xp = xb + (size_t)(c >> 1) * 2 * MDIM;  // pair block
    v2f b0 = *(const v2f*)(xp +  0);                     // cols +0,+16,+32,+48
    v2f b1 = *(const v2f*)(xp + 32);
    v2f b2 = *(const v2f*)(xp + 64);
    v2f b3 = *(const v2f*)(xp + 96);
    acc0 = __builtin_amdgcn_wmma_f32_16x16x4_f32(false, a, false, b0, (short)0, acc0, false, false);
    acc1 = __builtin_amdgcn_wmma_f32_16x16x4_f32(false, a, false, b1, (short)0, acc1, false, false);
    acc2 = __builtin_amdgcn_wmma_f32_16x16x4_f32(false, a, false, b2, (short)0, acc2, false, false);
    acc3 = __builtin_amdgcn_wmma_f32_16x16x4_f32(false, a, false, b3, (short)0, acc3, false, false);
  }

  float* yb = Y + (size_t)(ot * 16) * MDIM + (size_t)ms * 64 + l4;
#pragma unroll
  for (int r = 0; r < 8; ++r) {
    float* yr = yb + (size_t)(r + (hi ? 8 : 0)) * MDIM;
    yr[0]  = acc0[r];
    yr[16] = acc1[r];
    yr[32] = acc2[r];
    yr[48] = acc3[r];
  }
}

// ---------------------------------------------------------------------------
// 4) Global BN stats: block per channel, reduce sum/sumsq over M columns.
//    (Y is channel-major, rows fully coalesced.)
// ---------------------------------------------------------------------------
__global__ void k_bnstats(const float* __restrict__ Y, float* __restrict__ mean,
                          float* __restrict__ rstd) {
  int ch = blockIdx.x;
  const float* y = Y + (size_t)ch * MDIM;
  float s = 0.f, sq = 0.f;
  for (int m = threadIdx.x; m < MDIM; m += blockDim.x) {
    float v = y[m];
    s += v; sq += v * v;
  }
  __shared__ float sh[256], sh2[256];
  sh[threadIdx.x] = s; sh2[threadIdx.x] = sq;
  __syncthreads();
  for (int o = blockDim.x >> 1; o > 0; o >>= 1) {
    if ((int)threadIdx.x < o) {
      sh[threadIdx.x]  += sh[threadIdx.x + o];
      sh2[threadIdx.x] += sh2[threadIdx.x + o];
    }
    __syncthreads();
  }
  if (threadIdx.x == 0) {
    float mu = sh[0] / (float)MDIM;
    float var = sh2[0] / (float)MDIM - mu * mu;
    mean[ch] = mu;
    rstd[ch] = rsqrtf(var + EPSC);
  }
}

// ---------------------------------------------------------------------------
// 5) BN + ReLU apply: one thread per (channel-pair, m). Writes one float2
//    into the pair-interleaved next-layer X (if any) and two masked scalars
//    into the channel-major output slice (if any).
// ---------------------------------------------------------------------------
__global__ void k_apply(const float* __restrict__ Y, const float* __restrict__ mean,
                        const float* __restrict__ rstd, const float* __restrict__ g,
                        const float* __restrict__ beta, float* __restrict__ Xnext,
                        float* __restrict__ out, int outbase,
                        const float* __restrict__ validf) {
  int t = blockIdx.x * blockDim.x + threadIdx.x;
  int pr = t >> 17;            // channel pair
  int m  = t & (MDIM - 1);
  int c0 = pr * 2, c1 = c0 + 1;
  float v0 = (Y[(size_t)c0 * MDIM + m] - mean[c0]) * rstd[c0] * g[c0] + beta[c0];
  float v1 = (Y[(size_t)c1 * MDIM + m] - mean[c1]) * rstd[c1] * g[c1] + beta[c1];
  v0 = fmaxf(v0, 0.0f);
  v1 = fmaxf(v1, 0.0f);
  if (Xnext) {
    v2f w2 = {v0, v1};
    *(v2f*)(Xnext + (size_t)pr * 2 * MDIM + (size_t)m * 2) = w2;
  }
  if (out) {
    int b = m >> 13, rem = m & 8191, p = rem >> 5;
    float vm = validf[(b << 8) + p];
    float* ob = out + (size_t)(b * 640 + outbase) * 8192 + rem;
    ob[(size_t)c0 * 8192] = v0 * vm;
    ob[(size_t)c1 * 8192] = v1 * vm;
  }
}

// ---------------------------------------------------------------------------
extern "C" void kernel_launch(void* const* d_in, const int* in_sizes, int n_in,
                              void* d_out, int out_size, void* d_ws, size_t ws_size,
                              hipStream_t stream) {
  (void)in_sizes; (void)n_in; (void)out_size; (void)ws_size;
  const float* pc     = (const float*)d_in[0];
  const float* feat   = (const float*)d_in[1];
  const float* img1   = (const float*)d_in[2];
  const float* img2   = (const float*)d_in[3];
  // d_in[4] = P (unused by reference)
  const int*   qv1    = (const int*)d_in[5];
  const float* new_pc = (const float*)d_in[6];
  const float* w1 = (const float*)d_in[7];
  const float* g1 = (const float*)d_in[8];
  const float* b1 = (const float*)d_in[9];
  const float* w2 = (const float*)d_in[10];
  const float* g2 = (const float*)d_in[11];
  const float* b2 = (const float*)d_in[12];
  const float* w3 = (const float*)d_in[13];
  const float* g3 = (const float*)d_in[14];
  const float* b3 = (const float*)d_in[15];
  const float* w4 = (const float*)d_in[16];
  const float* g4 = (const float*)d_in[17];
  const float* b4 = (const float*)d_in[18];
  float* out = (float*)d_out;

  // workspace partition (aligned)
  char* ws = (char*)d_ws;
  size_t off = 0;
  int*   idx    = (int*)(ws + off);   off += (size_t)MDIM * 4;           // 512 KB
  float* validf = (float*)(ws + off); off += (size_t)BB * PCTR * 4;      // 16 KB
  off = (off + 255) & ~(size_t)255;
  float* stats  = (float*)(ws + off); off += 4 * 512 * 4;                // mean+rstd x4 layers
  off = (off + 255) & ~(size_t)255;
  float* w1pad  = (float*)(ws + off); off += (size_t)64 * 68 * 4;        // padded w1
  off = (off + 255) & ~(size_t)255;
  float* bufA   = (float*)(ws + off); off += (size_t)128 * MDIM * 4;     // X buffer (<=128 ch)
  float* bufB   = (float*)(ws + off); off += (size_t)256 * MDIM * 4;     // Y buffer (<=256 ch)

  const int strips_m = MDIM / 64;  // 2048 (16x64 output strips per Cout tile)

  k_query<<<(BB * PCTR + 255) / 256, 256, 0, stream>>>(pc, new_pc, idx, validf);
  k_padw1<<<(64 * 68 + 255) / 256, 256, 0, stream>>>(w1, w1pad);
  k_gather<<<MDIM / 256, 256, 0, stream>>>(pc, feat, img1, img2, qv1, new_pc,
                                           idx, validf, bufA, out);
  // Layer 1: 64 <- 68(padded)
  k_gemm<68><<<(64 / 16) * strips_m / 8, 256, 0, stream>>>(w1pad, bufA, bufB);
  k_bnstats<<<64, 256, 0, stream>>>(bufB, stats + 0, stats + 256);
  k_apply<<<(64 / 2) * (MDIM / 256), 256, 0, stream>>>(bufB, stats + 0, stats + 256,
                                                       g1, b1, bufA, out, 0, validf);
  // Layer 2: 128 <- 64
  k_gemm<64><<<(128 / 16) * strips_m / 8, 256, 0, stream>>>(w2, bufA, bufB);
  k_bnstats<<<128, 256, 0, stream>>>(bufB, stats + 512, stats + 768);
  k_apply<<<(128 / 2) * (MDIM / 256), 256, 0, stream>>>(bufB, stats + 512, stats + 768,
                                                        g2, b2, bufA, out, 128, validf);
  // Layer 3: 128 <- 128 (feeds layer 4 only)
  k_gemm<128><<<(128 / 16) * strips_m / 8, 256, 0, stream>>>(w3, bufA, bufB);
  k_bnstats<<<128, 256, 0, stream>>>(bufB, stats + 1024, stats + 1280);
  k_apply<<<(128 / 2) * (MDIM / 256), 256, 0, stream>>>(bufB, stats + 1024, stats + 1280,
                                                        g3, b3, bufA, nullptr, 0, validf);
  // Layer 4: 256 <- 128 (output only, channels 384..639)
  k_gemm<128><<<(256 / 16) * strips_m / 8, 256, 0, stream>>>(w4, bufA, bufB);
  k_bnstats<<<256, 256, 0, stream>>>(bufB, stats + 1536, stats + 1792);
  k_apply<<<(256 / 2) * (MDIM / 256), 256, 0, stream>>>(bufB, stats + 1536, stats + 1792,
                                                        g4, b4, nullptr, out, 384, validf);
}